// Transformer_34059090657387
// MI455X (gfx1250) — compile-verified
//
#include <hip/hip_runtime.h>
#include <math.h>

// ---------------- problem constants ----------------
#define BEAM    3
#define MAX_GEN 40
#define SEQ     384
#define VOCAB   30522
#define HIDDEN  768
#define LTOT    (SEQ + MAX_GEN)   // 424
#define PAD     0
#define SEP     102
#define MASK_T  103
#define QMARK   1029

#define NB    64                      // blocks for the GEMV kernel
#define NT    256                     // threads per block (8 waves of 32)
#define NW    8                       // waves per block
#define TPW   4                       // column tiles (of 16) per wave
#define NTILE ((VOCAB + 15) / 16)     // 1908 column tiles (64*8*4 = 2048 slots)

typedef __attribute__((ext_vector_type(2))) float v2f;
typedef __attribute__((ext_vector_type(8))) float v8f;

#define NEG_BIG (-1.0e30f)

// insert (v,ix) into descending top-3 list; strict > keeps earliest on ties
__device__ __forceinline__ void ins3(float v, int ix, float* bv, int* bi) {
    if (v > bv[0]) {
        bv[2] = bv[1]; bi[2] = bi[1];
        bv[1] = bv[0]; bi[1] = bi[0];
        bv[0] = v;     bi[0] = ix;
    } else if (v > bv[1]) {
        bv[2] = bv[1]; bi[2] = bi[1];
        bv[1] = v;     bi[1] = ix;
    } else if (v > bv[2]) {
        bv[2] = v;     bi[2] = ix;
    }
}

// ---------------- init state ----------------
__global__ void k_init(const int* __restrict__ tokens, int* beams, float* probs,
                       int* gen, int* len_p, int* ite_p) {
    int tid = blockIdx.x * blockDim.x + threadIdx.x;
    int stride = gridDim.x * blockDim.x;
    for (int i = tid; i < BEAM * LTOT; i += stride) {
        int pos = i % LTOT;
        beams[i] = (pos < SEQ) ? tokens[pos] : PAD;
    }
    if (tid < BEAM) { probs[tid] = (tid == 0) ? 1.0f : 0.0f; gen[tid] = 1; }
    if (tid == 0)   { *len_p = SEQ; *ite_p = 0; }
}

// ---------------- per-step: WMMA GEMV + partial softmax-sum + partial top3 ----------------
__global__ void __launch_bounds__(NT)
k_gemv_topk(const float* __restrict__ emb, const float* __restrict__ W,
            const float* __restrict__ bias, const int* __restrict__ beams,
            const int* __restrict__ len_p,
            float* __restrict__ pt_val, int* __restrict__ pt_idx,
            float* __restrict__ p_sum) {
    __shared__ float hid[BEAM * HIDDEN];   // 9 KB: hidden rows for the 3 beams
    __shared__ float s_val[BEAM * NT * 3];
    __shared__ int   s_idx[BEAM * NT * 3];
    __shared__ float s_se [BEAM * NT];

    const int tid = threadIdx.x;
    const int len = *len_p;

    // gather hidden = emb_table[last_token] for each beam (rows stay hot in L2)
    for (int i = tid; i < BEAM * HIDDEN; i += NT) {
        int b = i / HIDDEN, k = i % HIDDEN;
        int tok = beams[b * LTOT + (len - 1)];
        hid[i] = emb[(size_t)tok * HIDDEN + k];
    }
    __syncthreads();

    const int lane  = tid & 31;
    const int wave  = tid >> 5;
    const int gwave = blockIdx.x * NW + wave;
    const int m     = lane & 15;            // N column within tile / M row id
    const bool lo   = lane < 16;            // K phase select for 16x16x4 layout
    const int mc    = (m < BEAM) ? m : 0;   // rows >= BEAM of D are never read

    // Per-lane clamped column addresses for the wave's 4 tiles (unconditional loads;
    // out-of-range columns are duplicates, excluded in the epilogue).
    const int t0 = gwave * TPW;
    int   colv[TPW];
    const float* pW[TPW];
#pragma unroll
    for (int tt = 0; tt < TPW; ++tt) {
        int col = (t0 + tt) * 16 + m;
        colv[tt] = col;
        int colc = (col < VOCAB) ? col : (VOCAB - 1);
        pW[tt] = W + colc;
    }
    const float* ha = hid + mc * HIDDEN + (lo ? 0 : 2);

    v8f c[TPW];
#pragma unroll
    for (int tt = 0; tt < TPW; ++tt) c[tt] = (v8f){};

    // K loop: per chunk, 1 shared A slice feeds 4 independent WMMAs.
    for (int k0 = 0; k0 < HIDDEN; k0 += 4) {
        v2f a;
        a.x = ha[k0];
        a.y = ha[k0 + 1];
        v2f b[TPW];
#pragma unroll
        for (int tt = 0; tt < TPW; ++tt) {
            b[tt].x = pW[tt][(size_t)(k0 + (lo ? 0 : 2)) * VOCAB];
            b[tt].y = pW[tt][(size_t)(k0 + (lo ? 1 : 3)) * VOCAB];
        }
#pragma unroll
        for (int tt = 0; tt < TPW; ++tt) {
            c[tt] = __builtin_amdgcn_wmma_f32_16x16x4_f32(
                        false, a, false, b[tt], (short)0, c[tt], false, false);
        }
    }

    // epilogue: rows 0..2 of each tile are the beam logits for column colv[tt]
    float bv[BEAM][3]; int bi[BEAM][3]; float se[BEAM];
#pragma unroll
    for (int r = 0; r < BEAM; ++r) {
        se[r] = 0.0f;
#pragma unroll
        for (int j = 0; j < 3; ++j) { bv[r][j] = NEG_BIG; bi[r][j] = PAD; }
    }
#pragma unroll
    for (int tt = 0; tt < TPW; ++tt) {
        const int col = colv[tt];
        if (lo && col < VOCAB) {
            const float bb = bias[col];
            float l0 = c[tt][0] + bb, l1 = c[tt][1] + bb, l2 = c[tt][2] + bb;
            se[0] += expf(l0); ins3(l0, col, bv[0], bi[0]);
            se[1] += expf(l1); ins3(l1, col, bv[1], bi[1]);
            se[2] += expf(l2); ins3(l2, col, bv[2], bi[2]);
        }
    }

    // stash per-thread results
#pragma unroll
    for (int r = 0; r < BEAM; ++r) {
        s_se[r * NT + tid] = se[r];
#pragma unroll
        for (int j = 0; j < 3; ++j) {
            s_val[(r * NT + tid) * 3 + j] = bv[r][j];
            s_idx[(r * NT + tid) * 3 + j] = bi[r][j];
        }
    }
    __syncthreads();

    // deterministic block-level merge: thread r handles beam r
    if (tid < BEAM) {
        const int r = tid;
        float sum = 0.0f;
        for (int t2 = 0; t2 < NT; ++t2) sum += s_se[r * NT + t2];
        float mv[3] = {NEG_BIG, NEG_BIG, NEG_BIG}; int mi[3] = {PAD, PAD, PAD};
        for (int t2 = 0; t2 < NT; ++t2)
            for (int j = 0; j < 3; ++j)
                ins3(s_val[(r * NT + t2) * 3 + j], s_idx[(r * NT + t2) * 3 + j], mv, mi);
        for (int j = 0; j < 3; ++j) {
            pt_val[blockIdx.x * 9 + r * 3 + j] = mv[j];
            pt_idx[blockIdx.x * 9 + r * 3 + j] = mi[j];
        }
        p_sum[blockIdx.x * BEAM + r] = sum;
    }
}

// ---------------- per-step: merge partials + beam update (single block) ----------------
__global__ void k_update(const float* __restrict__ pt_val, const int* __restrict__ pt_idx,
                         const float* __restrict__ p_sum,
                         int* beams, int* tmpb, float* probs, int* gen,
                         int* len_p, int* ite_p) {
    __shared__ float top_p[BEAM][3]; __shared__ int top_i[BEAM][3];
    __shared__ float top_bp[BEAM];   __shared__ int parent[BEAM]; __shared__ int newtok[BEAM];
    __shared__ int s_active;
    __shared__ int sepcnt[BEAM], qcnt[BEAM];

    const int tid = threadIdx.x;
    if (tid == 0) s_active = (gen[0] | gen[1] | gen[2]);
    __syncthreads();
    if (!s_active) return;                 // uniform: nothing changes this step

    const int length = *len_p;

    if (tid < BEAM) {                      // per-beam global merge + softmax values
        const int r = tid;
        float se = 0.0f;
        for (int b2 = 0; b2 < NB; ++b2) se += p_sum[b2 * BEAM + r];
        float mv[3] = {NEG_BIG, NEG_BIG, NEG_BIG}; int mi[3] = {PAD, PAD, PAD};
        for (int b2 = 0; b2 < NB; ++b2)
            for (int j = 0; j < 3; ++j)
                ins3(pt_val[b2 * 9 + r * 3 + j], pt_idx[b2 * 9 + r * 3 + j], mv, mi);
        for (int j = 0; j < 3; ++j) {
            top_p[r][j] = expf(mv[j]) / se;   // softmax probability of top-j token
            top_i[r][j] = mi[j];
        }
    }
    __syncthreads();

    if (tid == 0) {                        // 9-candidate stable top-3 (JAX semantics)
        float cand_p[9]; int cand_t[9];
        for (int b = 0; b < BEAM; ++b)
            for (int j = 0; j < 3; ++j) {
                int cc = b * 3 + j;
                if (gen[b]) { cand_p[cc] = probs[b] * top_p[b][j]; cand_t[cc] = top_i[b][j]; }
                else        { cand_p[cc] = (j == 0) ? probs[b] : 0.0f; cand_t[cc] = PAD; }
            }
        bool used[9] = {false,false,false,false,false,false,false,false,false};
        for (int sel = 0; sel < BEAM; ++sel) {
            float best = NEG_BIG * 2.0f; int bj = 0;
            for (int j = 0; j < 9; ++j)
                if (!used[j] && cand_p[j] > best) { best = cand_p[j]; bj = j; }
            used[bj] = true;
            top_bp[sel] = cand_p[bj];
            parent[sel] = bj / 3;
            newtok[sel] = cand_t[bj];
        }
    }
    __syncthreads();

    // gather parent rows into tmp, appending new token at position `length`
    for (int i = tid; i < BEAM * LTOT; i += blockDim.x) {
        int row = i / LTOT, pos = i % LTOT;
        int v = beams[parent[row] * LTOT + pos];
        if (pos == length) v = newtok[row];
        tmpb[i] = v;
    }
    __syncthreads();
    if (tid < BEAM) { sepcnt[tid] = 0; qcnt[tid] = 0; }
    __syncthreads();
    // write back + EOS counting (integer shared atomics: deterministic)
    for (int i = tid; i < BEAM * LTOT; i += blockDim.x) {
        int v = tmpb[i];
        beams[i] = v;
        int row = i / LTOT;
        if (v == SEP)   atomicAdd(&sepcnt[row], 1);
        if (v == QMARK) atomicAdd(&qcnt[row], 1);
    }
    __syncthreads();
    if (tid < BEAM) {
        probs[tid] = top_bp[tid];
        gen[tid]   = (sepcnt[tid] <= 3) && (qcnt[tid] == 0);
    }
    if (tid == 0) { *len_p = length + 1; *ite_p = *ite_p + 1; }
}

// ---------------- final output: (generated[40], probs[3], ite) as floats ----------------
__global__ void k_out(const int* __restrict__ beams, const float* __restrict__ probs,
                      const int* __restrict__ ite_p, float* out, int out_size) {
    __shared__ int s_best;
    const int ite = *ite_p;
    if (threadIdx.x == 0) {
        int best = 0; float bp = probs[0];
        for (int b = 1; b < BEAM; ++b) if (probs[b] > bp) { bp = probs[b]; best = b; }
        s_best = best;
    }
    __syncthreads();
    const int best = s_best;
    for (int j = threadIdx.x; j < MAX_GEN; j += blockDim.x) {
        int tok = beams[best * LTOT + SEQ + j];
        float v = ((j < ite) && (tok != MASK_T)) ? (float)tok : (float)PAD;
        if (j < out_size) out[j] = v;
    }
    if (threadIdx.x < BEAM && (MAX_GEN + threadIdx.x) < out_size)
        out[MAX_GEN + threadIdx.x] = probs[threadIdx.x];
    if (threadIdx.x == 0 && (MAX_GEN + BEAM) < out_size)
        out[MAX_GEN + BEAM] = (float)ite;
}

// ---------------- host ----------------
extern "C" void kernel_launch(void* const* d_in, const int* in_sizes, int n_in,
                              void* d_out, int out_size, void* d_ws, size_t ws_size,
                              hipStream_t stream) {
    const int*   tokens = (const int*)  d_in[0];   // [1,384]
    const float* emb    = (const float*)d_in[1];   // [30522,768]
    const float* W      = (const float*)d_in[2];   // [768,30522]
    const float* bias   = (const float*)d_in[3];   // [1,30522]

    char* w = (char*)d_ws;
    int*   beams  = (int*)w;    w += BEAM * LTOT * sizeof(int);
    int*   tmpb   = (int*)w;    w += BEAM * LTOT * sizeof(int);
    float* probs  = (float*)w;  w += 4 * sizeof(float);
    int*   gen    = (int*)w;    w += 4 * sizeof(int);
    int*   len_p  = (int*)w;    w += sizeof(int);
    int*   ite_p  = (int*)w;    w += 3 * sizeof(int);
    float* pt_val = (float*)w;  w += NB * 9 * sizeof(float);
    int*   pt_idx = (int*)w;    w += NB * 9 * sizeof(int);
    float* p_sum  = (float*)w;  w += NB * BEAM * sizeof(float);

    k_init<<<8, 256, 0, stream>>>(tokens, beams, probs, gen, len_p, ite_p);

    for (int s = 0; s < MAX_GEN; ++s) {
        k_gemv_topk<<<NB, NT, 0, stream>>>(emb, W, bias, beams, len_p,
                                           pt_val, pt_idx, p_sum);
        k_update<<<1, 128, 0, stream>>>(pt_val, pt_idx, p_sum,
                                        beams, tmpb, probs, gen, len_p, ite_p);
    }

    k_out<<<1, 64, 0, stream>>>(beams, probs, ite_p, (float*)d_out, out_size);
}